// SAW_8675833938580
// MI455X (gfx1250) — compile-verified
//
#include <hip/hip_runtime.h>
#include <hip/hip_bf16.h>
#include <math.h>

typedef __attribute__((ext_vector_type(2))) float v2f;
typedef __attribute__((ext_vector_type(4))) float v4f;
typedef __attribute__((ext_vector_type(8))) float v8f;

#define NCLS   19
#define CH     512
#define CSEL   16        // C
#define GRP    32        // CH / CSEL
#define BATCH  8
#define HW     16384     // 128*128
#define SPLIT  8         // K-splits per (b,g)

// Per-wave chunk: HW/SPLIT/8 = 256 elements; processed 32 per pipelined
// iteration (4 x b128 loads -> 8 WMMAs), 8 iterations total.
#define CHUNK  (HW / SPLIT / 8)      // 256
#define STEP   32                    // elements per pipeline stage
#define NIT    (CHUNK / STEP)        // 8

// Workspace layout:
//   [0,    2048)  int   chan[GRP*CSEL]
//   [2048, 4096)  float wgh [GRP*CSEL]
//   [4096, ... )  float cov [B*G*256]  raw (unscaled) Gram accumulators

// ---------------------------------------------------------------------------
// Kernel 1: stable top-32 selection of |w| per selected class.
// ---------------------------------------------------------------------------
__global__ void SAW_topk_setup(const float* __restrict__ w,
                               int* __restrict__ chan,
                               float* __restrict__ wgh) {
    int j = threadIdx.x;
    if (j >= CSEL) return;
    const float* wr = w + (size_t)j * CH;   // SELECTED_CLASSES = 0..15
    float pv = 3.4e38f;
    int   pc = -1;
    for (int k = 0; k < GRP; ++k) {
        float bv = -1.0f; int bc = 0;
        for (int c = 0; c < CH; ++c) {
            float v = fabsf(wr[c]);
            bool elig = (v < pv) || (v == pv && c > pc);
            if (elig && v > bv) { bv = v; bc = c; }
        }
        pv = bv; pc = bc;
        chan[k * CSEL + j] = bc;
        wgh[k * CSEL + j] = 1.0f / (1.0f + expf(-bv));
    }
}

// ---------------------------------------------------------------------------
// Kernel 2: zero the Gram accumulators.
// ---------------------------------------------------------------------------
__global__ void SAW_zero_cov(float* __restrict__ cov, int n) {
    int i = blockIdx.x * blockDim.x + threadIdx.x;
    if (i < n) cov[i] = 0.0f;
}

// ---------------------------------------------------------------------------
// Kernel 3: streaming Gram via V_WMMA_F32_16X16X4_F32.
// Each lane loads b128 (16 B) at element offset k + 4*half:
//   lo lanes: row elements k..k+3 ; hi lanes: row elements k+4..k+7.
// WMMA step 1 uses .xy of all lanes (K slots {0,1}=lo pair, {2,3}=hi pair),
// WMMA step 2 uses .zw. Gram sums over all HW positions, so this K-slot
// permutation is exact. One b128 feeds two WMMAs; tile covered exactly once.
// ---------------------------------------------------------------------------
__global__ void SAW_gram_wmma(const float* __restrict__ x,
                              const int* __restrict__ chan,
                              float* __restrict__ cov) {
    __shared__ float scov[CSEL * CSEL];

    int blk  = blockIdx.x;
    int b    = blk / (GRP * SPLIT);
    int rem  = blk % (GRP * SPLIT);
    int g    = rem / SPLIT;
    int s    = rem % SPLIT;
    int tid  = threadIdx.x;          // 256 threads = 8 waves
    int wave = tid >> 5;
    int lane = tid & 31;
    int row  = lane & 15;            // M index this lane feeds
    int half = lane >> 4;

    scov[tid] = 0.0f;
    __syncthreads();

    int c = chan[g * CSEL + row];
    const float* rp = x + ((size_t)b * CH + c) * (size_t)HW
                        + (size_t)(s * (HW / SPLIT) + wave * CHUNK)
                        + 4 * half;

    v8f acc = {0.f, 0.f, 0.f, 0.f, 0.f, 0.f, 0.f, 0.f};

#define LOADQ(off) (*(const v4f*)(rp + (off)))
#define WMMA2(q)                                                            \
    do {                                                                    \
        v2f lo_ = __builtin_shufflevector((q), (q), 0, 1);                  \
        v2f hi_ = __builtin_shufflevector((q), (q), 2, 3);                  \
        acc = __builtin_amdgcn_wmma_f32_16x16x4_f32(                        \
                  false, lo_, false, lo_, (short)0, acc, false, false);     \
        acc = __builtin_amdgcn_wmma_f32_16x16x4_f32(                        \
                  false, hi_, false, hi_, (short)0, acc, false, false);     \
    } while (0)

    // Software pipeline: 4 b128 loads in flight while 8 WMMAs consume the
    // previous group.
    v4f a0 = LOADQ(0), a1 = LOADQ(8), a2 = LOADQ(16), a3 = LOADQ(24);
    int k = STEP;
    #pragma unroll 1
    for (int it = 0; it < NIT - 1; ++it) {
        v4f n0 = LOADQ(k + 0), n1 = LOADQ(k + 8),
            n2 = LOADQ(k + 16), n3 = LOADQ(k + 24);
        WMMA2(a0); WMMA2(a1); WMMA2(a2); WMMA2(a3);
        a0 = n0; a1 = n1; a2 = n2; a3 = n3;
        k += STEP;
    }
    WMMA2(a0); WMMA2(a1); WMMA2(a2); WMMA2(a3);

#undef WMMA2
#undef LOADQ

    // C/D layout: VGPR r holds (M = r + 8*half, N = row). Reduce 8 waves.
    #pragma unroll
    for (int r = 0; r < 8; ++r) {
        atomicAdd(&scov[(r + 8 * half) * CSEL + row], acc[r]);   // ds_add_f32
    }
    __syncthreads();

    // One global fp32 atomic per entry per block (SPLIT blocks contend).
    unsafeAtomicAdd(&cov[((size_t)(b * GRP + g)) * 256 + tid], scov[tid]);
}

// ---------------------------------------------------------------------------
// Kernel 4: fold in sigmoid weights, masked |off-diag| sum, margin, clamp,
// batch-mean. One thread per (b,g) pair.
// ---------------------------------------------------------------------------
__global__ void SAW_loss_reduce(const float* __restrict__ cov,
                                const float* __restrict__ wgh,
                                float* __restrict__ out) {
    __shared__ float red[256];
    int t = threadIdx.x;             // pair index = b*GRP + g = t
    int g = t % GRP;
    const float* Gm = cov + (size_t)t * 256;

    float wloc[CSEL];
    #pragma unroll
    for (int j = 0; j < CSEL; ++j) wloc[j] = wgh[g * CSEL + j];

    float s = 0.0f;
    for (int c = 0; c < CSEL; ++c)
        for (int d = c + 1; d < CSEL; ++d)
            s += fabsf(Gm[c * CSEL + d]) * wloc[c] * wloc[d];

    s *= (1.0f / (float)(HW - 1));
    // num_off = 120 ; margin = floor(120/2) = 60
    float loss = fmaxf((s - 60.0f) * (1.0f / 120.0f), 0.0f);

    red[t] = loss;
    __syncthreads();
    for (int st = 128; st > 0; st >>= 1) {
        if (t < st) red[t] += red[t + st];
        __syncthreads();
    }
    if (t == 0) out[0] = red[0] / (float)BATCH;
}

// ---------------------------------------------------------------------------
extern "C" void kernel_launch(void* const* d_in, const int* in_sizes, int n_in,
                              void* d_out, int out_size, void* d_ws, size_t ws_size,
                              hipStream_t stream) {
    const float* x  = (const float*)d_in[0];   // [8,512,128,128] f32
    const float* cw = (const float*)d_in[1];   // [19,512] f32
    float* out = (float*)d_out;

    int*   chan = (int*)d_ws;
    float* wgh  = (float*)((char*)d_ws + 2048);
    float* cov  = (float*)((char*)d_ws + 4096);
    const int ncov = BATCH * GRP * CSEL * CSEL;   // 65536

    SAW_topk_setup<<<1, 16, 0, stream>>>(cw, chan, wgh);
    SAW_zero_cov<<<(ncov + 255) / 256, 256, 0, stream>>>(cov, ncov);
    SAW_gram_wmma<<<BATCH * GRP * SPLIT, 256, 0, stream>>>(x, chan, cov);
    SAW_loss_reduce<<<1, 256, 0, stream>>>(cov, wgh, out);
}